// CRPSLoss_12343736009041
// MI455X (gfx1250) — compile-verified
//
#include <hip/hip_runtime.h>
#include <hip/hip_bf16.h>

typedef __attribute__((ext_vector_type(2))) float v2f;
typedef __attribute__((ext_vector_type(8))) float v8f;

#define NEG_INF (-__builtin_inff())

// ---------------------------------------------------------------------------
// Kernel 1: one wave32 per row. Streaming b128 loads, register-resident row,
// wave shuffle reductions + prefix scan, fused (cdf - step)^2 accumulation.
// Writes one float partial per row to d_ws (deterministic).
// ---------------------------------------------------------------------------
__global__ __launch_bounds__(256)
void crps_rows_kernel(const float* __restrict__ logits,
                      const int*   __restrict__ labels,
                      float*       __restrict__ rowsum,
                      int B, int C) {
    const int lane = threadIdx.x & 31;
    const int wave = threadIdx.x >> 5;
    const int row  = blockIdx.x * 8 + wave;
    if (row >= B) return;                       // wave-uniform

    const float* rp    = logits + (size_t)row * (size_t)C;
    const int    label = labels[row];

    // Load up to 1024 elements: chunk c covers [c*128, c*128+128),
    // lane owns 4 consecutive floats -> global_load_b128 (16B aligned:
    // row stride 4000B is a multiple of 16, C % 4 == 0).
    float4 v[8];
#pragma unroll
    for (int c = 0; c < 8; ++c) {
        const int idx = c * 128 + lane * 4;
        if (idx < C) {
            v[c] = *(const float4*)(rp + idx);
        } else {
            v[c] = make_float4(NEG_INF, NEG_INF, NEG_INF, NEG_INF);
        }
    }

    // Row max (for numerically stable softmax).
    float m = NEG_INF;
#pragma unroll
    for (int c = 0; c < 8; ++c) {
        m = fmaxf(m, fmaxf(fmaxf(v[c].x, v[c].y), fmaxf(v[c].z, v[c].w)));
    }
#pragma unroll
    for (int off = 16; off >= 1; off >>= 1)
        m = fmaxf(m, __shfl_xor(m, off, 32));

    // exp(x - m) in place (padded lanes become exactly 0), plus row sum Z.
    float s = 0.0f;
#pragma unroll
    for (int c = 0; c < 8; ++c) {
        v[c].x = __expf(v[c].x - m);
        v[c].y = __expf(v[c].y - m);
        v[c].z = __expf(v[c].z - m);
        v[c].w = __expf(v[c].w - m);
        s += (v[c].x + v[c].y) + (v[c].z + v[c].w);
    }
#pragma unroll
    for (int off = 16; off >= 1; off >>= 1)
        s += __shfl_xor(s, off, 32);
    const float invZ = 1.0f / s;

    // Prefix scan across the row + fused squared-error accumulation.
    float carry = 0.0f;
    float acc   = 0.0f;
#pragma unroll
    for (int c = 0; c < 8; ++c) {
        const float p0 = v[c].x;
        const float p1 = p0 + v[c].y;
        const float p2 = p1 + v[c].z;
        const float p3 = p2 + v[c].w;

        // Inclusive shuffle scan of lane totals.
        float t = p3;
#pragma unroll
        for (int off = 1; off < 32; off <<= 1) {
            const float nbr = __shfl_up(t, off, 32);
            if (lane >= off) t += nbr;
        }
        const float excl = t - p3;               // exclusive prefix of lane totals
        const float base = carry + excl;
        const float ctot = __shfl(t, 31, 32);    // chunk total

        const int idx = c * 128 + lane * 4;
        if (idx < C) {
            float cdf, d;
            cdf = (base + p0) * invZ; d = cdf - (float)(idx + 0 >= label); acc = fmaf(d, d, acc);
            cdf = (base + p1) * invZ; d = cdf - (float)(idx + 1 >= label); acc = fmaf(d, d, acc);
            cdf = (base + p2) * invZ; d = cdf - (float)(idx + 2 >= label); acc = fmaf(d, d, acc);
            cdf = (base + p3) * invZ; d = cdf - (float)(idx + 3 >= label); acc = fmaf(d, d, acc);
        }
        carry += ctot;
    }

#pragma unroll
    for (int off = 16; off >= 1; off >>= 1)
        acc += __shfl_xor(acc, off, 32);
    if (lane == 0) rowsum[row] = acc;
}

// ---------------------------------------------------------------------------
// Kernel 2: deterministic final reduction via WMMA ones-matmul.
// D = Ones(16x4) x B(4x16) + C  =>  D[m,n] = sum_k B[k,n], identical rows.
// Summing VGPR0 of the accumulator across all 32 lanes (documented C/D
// layout: rows M=0 and M=8, N=0..15) yields exactly 2x the grand total,
// independent of the A/B operand lane mappings. Main body is guard-free
// (unconditional b64 loads, pipelineable); generic tail folded into the
// final shuffle reduction.
// ---------------------------------------------------------------------------
__global__ __launch_bounds__(32)
void crps_reduce_wmma_kernel(const float* __restrict__ rowsum,
                             float*       __restrict__ out,
                             int n, float scale) {
    const int lane = threadIdx.x & 31;

    v2f a; a.x = 1.0f; a.y = 1.0f;   // all-ones A matrix (layout independent)
    v8f c = {};

    const int nmain = n & ~63;       // 64 elements per WMMA step
    const float2* p2 = (const float2*)rowsum;
    for (int i = 0; i < nmain; i += 64) {
        const float2 d = p2[(i >> 1) + lane];   // elements i + 2*lane (+1)
        v2f b; b.x = d.x; b.y = d.y;
        // 8 args: (neg_a, A, neg_b, B, c_mod, C, reuse_a, reuse_b)
        c = __builtin_amdgcn_wmma_f32_16x16x4_f32(
                false, a, false, b, (short)0, c, false, false);
    }

    // Scalar tail (empty when n % 64 == 0, e.g. n = 16384).
    float tail = 0.0f;
    for (int k = nmain + lane; k < n; k += 32) tail += rowsum[k];

    // c[0] summed over 32 lanes double-counts (rows M=0 and M=8 identical).
    float t = c[0] * 0.5f + tail;
#pragma unroll
    for (int off = 16; off >= 1; off >>= 1)
        t += __shfl_xor(t, off, 32);

    if (lane == 0) out[0] = t * scale;
}

// ---------------------------------------------------------------------------
extern "C" void kernel_launch(void* const* d_in, const int* in_sizes, int n_in,
                              void* d_out, int out_size, void* d_ws, size_t ws_size,
                              hipStream_t stream) {
    const float* logits = (const float*)d_in[0];
    const int*   labels = (const int*)d_in[1];

    const int B = in_sizes[1];                 // 16384
    const int C = in_sizes[0] / B;             // 1000

    float* rowsum = (float*)d_ws;              // B floats of scratch

    dim3 grid((B + 7) / 8);
    crps_rows_kernel<<<grid, 256, 0, stream>>>(logits, labels, rowsum, B, C);

    const float scale = 1.0f / ((float)B * (float)C);
    crps_reduce_wmma_kernel<<<1, 32, 0, stream>>>(rowsum, (float*)d_out, B, scale);
}